// CoAttention_73710228734830
// MI455X (gfx1250) — compile-verified
//
#include <hip/hip_runtime.h>
#include <hip/hip_bf16.h>

typedef __attribute__((ext_vector_type(2))) float v2f;
typedef __attribute__((ext_vector_type(8))) float v8f;

#define EPS 1e-3f

// ---- problem constants (match reference setup_inputs) ----
#define CB 512
#define CN 100
#define CT 14
#define CD 2048
#define CH 768

__device__ __forceinline__ v8f wmma_f32(v2f a, v2f b, v8f c) {
    return __builtin_amdgcn_wmma_f32_16x16x4_f32(
        /*neg_a=*/false, a, /*neg_b=*/false, b,
        /*c_mod=*/(short)0, c, /*reuse_a=*/false, /*reuse_b=*/false);
}

__device__ __forceinline__ float wredsum(float v) {
    #pragma unroll
    for (int m = 16; m >= 1; m >>= 1) v += __shfl_xor(v, m, 32);
    return v;
}
__device__ __forceinline__ float wredmax(float v) {
    #pragma unroll
    for (int m = 16; m >= 1; m >>= 1) v = fmaxf(v, __shfl_xor(v, m, 32));
    return v;
}

// =====================================================================
// Kernel 1: C1 = A @ W1 + b1  (and optionally C2 = A @ W2 + b2, sharing A)
// A:[M,K] row-major, W:[K,Hn] row-major, C:[M,Hn]. M%64==0, K%32==0, Hn%64==0.
// Block = 256 threads = 8 waves in 4x2 (M x N); wave tile = 16x32.
// =====================================================================
template <bool DUAL>
__global__ __launch_bounds__(256) void gemm_bias_kernel(
    const float* __restrict__ A,
    const float* __restrict__ W1, const float* __restrict__ b1v,
    const float* __restrict__ W2, const float* __restrict__ b2v,
    float* __restrict__ C1, float* __restrict__ C2,
    int M, int K, int Hn)
{
    __shared__ float As[64][33];     // +1 pad: odd stride -> conflict-free col reads
    __shared__ float Bs1[32][80];    // stride 80: halves of wave hit disjoint banks
    __shared__ float Bs2[32][80];

    const int tid  = threadIdx.x;
    const int w    = tid >> 5;
    const int lane = tid & 31;
    const int half = lane >> 4;
    const int lr   = lane & 15;

    const int m0 = blockIdx.x * 64;
    const int n0 = blockIdx.y * 64;

    const int wm_ = w >> 1;          // 0..3
    const int wn_ = w & 1;           // 0..1
    const int row0 = wm_ * 16;
    const int col0 = wn_ * 32;

    v8f acc1[2] = { (v8f){}, (v8f){} };
    v8f acc2[2] = { (v8f){}, (v8f){} };

    for (int k0 = 0; k0 < K; k0 += 32) {
        // stage A tile 64x32
        #pragma unroll
        for (int i = tid; i < 512; i += 256) {
            int r = i >> 3, cv = (i & 7) << 2;
            float4 v = *(const float4*)(A + (size_t)(m0 + r) * K + k0 + cv);
            As[r][cv + 0] = v.x; As[r][cv + 1] = v.y;
            As[r][cv + 2] = v.z; As[r][cv + 3] = v.w;
        }
        // stage B tiles 32x64
        #pragma unroll
        for (int i = tid; i < 512; i += 256) {
            int r = i >> 4, cv = (i & 15) << 2;
            float4 v1 = *(const float4*)(W1 + (size_t)(k0 + r) * Hn + n0 + cv);
            Bs1[r][cv + 0] = v1.x; Bs1[r][cv + 1] = v1.y;
            Bs1[r][cv + 2] = v1.z; Bs1[r][cv + 3] = v1.w;
            if (DUAL) {
                float4 v2 = *(const float4*)(W2 + (size_t)(k0 + r) * Hn + n0 + cv);
                Bs2[r][cv + 0] = v2.x; Bs2[r][cv + 1] = v2.y;
                Bs2[r][cv + 2] = v2.z; Bs2[r][cv + 3] = v2.w;
            }
        }
        __syncthreads();

        #pragma unroll
        for (int kk = 0; kk < 32; kk += 4) {
            int ka = kk + 2 * half;
            v2f a = { As[row0 + lr][ka], As[row0 + lr][ka + 1] };
            #pragma unroll
            for (int nt = 0; nt < 2; nt++) {
                int c = col0 + nt * 16 + lr;
                v2f b1f = { Bs1[ka][c], Bs1[ka + 1][c] };
                acc1[nt] = wmma_f32(a, b1f, acc1[nt]);
                if (DUAL) {
                    v2f b2f = { Bs2[ka][c], Bs2[ka + 1][c] };
                    acc2[nt] = wmma_f32(a, b2f, acc2[nt]);
                }
            }
        }
        __syncthreads();
    }

    // epilogue: bias + store. C-frag: VGPR i -> row i+8*half, col lr (per 16-tile)
    #pragma unroll
    for (int nt = 0; nt < 2; nt++) {
        int col = n0 + col0 + nt * 16 + lr;
        float bv1 = b1v[col];
        float bv2 = DUAL ? b2v[col] : 0.0f;
        #pragma unroll
        for (int i = 0; i < 8; i++) {
            int row = m0 + row0 + i + 8 * half;
            C1[(size_t)row * Hn + col] = acc1[nt][i] + bv1;
            if (DUAL) C2[(size_t)row * Hn + col] = acc2[nt][i] + bv2;
        }
    }
}

// =====================================================================
// Kernel 2: fused per-batch co-attention. One 256-thread block per batch.
// =====================================================================
__global__ __launch_bounds__(256) void coattn_fused_kernel(
    const float* __restrict__ img_feat,   // [B,N,D]
    const float* __restrict__ ques_feat,  // [B,T,H]
    const float* __restrict__ img_corr_g, // [B,N,H]
    const float* __restrict__ img_embed_g,// [B,N,H]
    const float* __restrict__ ques_embed_g,// [B,T,H]
    const float* __restrict__ wqa, const float* __restrict__ bqa,
    const float* __restrict__ wia, const float* __restrict__ bia,
    const float* __restrict__ g1, const float* __restrict__ b1,
    const float* __restrict__ g2, const float* __restrict__ b2,
    const float* __restrict__ g3, const float* __restrict__ b3,
    const float* __restrict__ g4, const float* __restrict__ b4,
    const float* __restrict__ g5, const float* __restrict__ b5,
    float* __restrict__ out_img,          // [B,D]
    float* __restrict__ out_q)            // [B,H]
{
    __shared__ float wm[16][113];     // padded [T=14->16][N=100->112(+1)]
    __shared__ float tile[16][CH];    // one 16-row x H staging tile
    __shared__ float s_qraw[16], s_qatt[16];
    __shared__ float s_iraw[112], s_iatt[112];

    const int b    = blockIdx.x;
    const int tid  = threadIdx.x;
    const int w    = tid >> 5;
    const int lane = tid & 31;
    const int half = lane >> 4;
    const int lr   = lane & 15;

    const float* qf  = ques_feat    + (size_t)b * CT * CH;
    const float* imf = img_feat     + (size_t)b * CN * CD;
    const float* ic  = img_corr_g   + (size_t)b * CN * CH;
    const float* ie  = img_embed_g  + (size_t)b * CN * CH;
    const float* qe  = ques_embed_g + (size_t)b * CT * CH;

    // ---- P1: wm = ques_feat(14x768) @ img_corr^T(768x100) via WMMA ------
    // 7 N-tiles of 16 -> waves 0..6. Branch-free lane masking: clamp the
    // row pointer (loop-invariant) and scale the fragment by 0/1.
    if (w < 7) {
        const int n0 = w * 16;
        const float am = (lr < CT) ? 1.0f : 0.0f;
        const float bm = ((n0 + lr) < CN) ? 1.0f : 0.0f;
        const int ta = (lr < CT) ? lr : 0;
        const int nb = ((n0 + lr) < CN) ? (n0 + lr) : 0;
        const float* pa = qf + (size_t)ta * CH;
        const float* pb = ic + (size_t)nb * CH;
        v8f acc = (v8f){};
        #pragma unroll 4
        for (int kk = 0; kk < CH; kk += 4) {
            int ka = kk + 2 * half;
            v2f a  = *(const v2f*)(pa + ka);
            v2f bf = *(const v2f*)(pb + ka);
            a *= am; bf *= bm;
            acc = wmma_f32(a, bf, acc);
        }
        #pragma unroll
        for (int i = 0; i < 8; i++) {
            int t = i + 8 * half;
            int n = n0 + lr;
            wm[t][n] = (t < CT && n < CN) ? acc[i] : 0.0f;  // zero-pad rows/cols
        }
    }
    __syncthreads();

    // ---- P2: wm = relu(LN_n(wm, g1, b1)) --------------------------------
    if (tid < CT) {
        float s = 0.f, sq = 0.f;
        for (int n = 0; n < CN; n++) { float v = wm[tid][n]; s += v; sq += v * v; }
        float mu = s / CN, var = sq / CN - mu * mu, rs = rsqrtf(var + EPS);
        for (int n = 0; n < CN; n++) {
            float v = (wm[tid][n] - mu) * rs * g1[n] + b1[n];
            wm[tid][n] = fmaxf(v, 0.0f);
        }
    }
    __syncthreads();

    // ---- P3: transform_img = wm(16x100) @ img_embed(100x768) via WMMA ---
    for (int ht = w; ht < CH / 16; ht += 8) {
        int h0 = ht * 16;
        v8f acc = (v8f){};
        #pragma unroll 5
        for (int kk = 0; kk < CN; kk += 4) {        // 100 % 4 == 0, exact
            int ka = kk + 2 * half;
            v2f a = { wm[lr][ka], wm[lr][ka + 1] };
            v2f bf = { ie[(size_t)ka * CH + h0 + lr],
                       ie[(size_t)(ka + 1) * CH + h0 + lr] };
            acc = wmma_f32(a, bf, acc);
        }
        #pragma unroll
        for (int i = 0; i < 8; i++) tile[i + 8 * half][h0 + lr] = acc[i];
    }
    __syncthreads();

    // ---- P4: qa_sum = relu(LN_h(timg + ques_embed)); qraw = qa_sum @ wqa
    for (int t = w; t < 16; t += 8) {
        if (t < CT) {
            float vv[CH / 32];
            float s = 0.f, sq = 0.f;
            #pragma unroll
            for (int i = 0; i < CH / 32; i++) {
                int h = lane + 32 * i;
                float v = tile[t][h] + qe[(size_t)t * CH + h];
                vv[i] = v; s += v; sq += v * v;
            }
            s = wredsum(s); sq = wredsum(sq);
            float mu = s / CH, var = sq / CH - mu * mu, rs = rsqrtf(var + EPS);
            float acc = 0.f;
            #pragma unroll
            for (int i = 0; i < CH / 32; i++) {
                int h = lane + 32 * i;
                float v = fmaxf((vv[i] - mu) * rs * g2[h] + b2[h], 0.0f);
                acc += v * wqa[h];
            }
            acc = wredsum(acc);
            if (lane == 0) s_qraw[t] = acc + bqa[0];
        }
    }
    __syncthreads();

    // ---- P5: ques_atten = softmax(LN_t(qraw, g3, b3)) (wave 0) ----------
    if (w == 0) {
        bool act = (lane < CT);
        float x = act ? s_qraw[lane] : 0.0f;
        float s = wredsum(x), sq = wredsum(x * x);
        float mu = s / CT, var = sq / CT - mu * mu, rs = rsqrtf(var + EPS);
        float y = act ? ((x - mu) * rs * g3[lane] + b3[lane]) : -1e30f;
        float m = wredmax(y);
        float e = act ? __expf(y - m) : 0.0f;
        float se = wredsum(e);
        if (act) s_qatt[lane] = e / se;
    }
    __syncthreads();

    // ---- P6: transform_ques = wm^T(100x14) @ ques_embed(14x768) per m-tile
    // K = T padded to 16: wm rows 14,15 are zero; B rows >= T are masked
    // branch-free via clamped row index + 0/1 multiply.
    for (int mt = 0; mt < 7; mt++) {
        const int m0 = mt * 16;
        for (int ht = w; ht < CH / 16; ht += 8) {
            int h0 = ht * 16;
            v8f acc = (v8f){};
            #pragma unroll
            for (int kk = 0; kk < 16; kk += 4) {
                int ka = kk + 2 * half;
                v2f a = { wm[ka][m0 + lr], wm[ka + 1][m0 + lr] }; // wm^T read
                int r0 = (ka     < CT) ? ka     : 0;
                int r1 = (ka + 1 < CT) ? ka + 1 : 0;
                float mk0 = (ka     < CT) ? 1.0f : 0.0f;
                float mk1 = (ka + 1 < CT) ? 1.0f : 0.0f;
                v2f bf = { qe[(size_t)r0 * CH + h0 + lr] * mk0,
                           qe[(size_t)r1 * CH + h0 + lr] * mk1 };
                acc = wmma_f32(a, bf, acc);
            }
            #pragma unroll
            for (int i = 0; i < 8; i++) tile[i + 8 * half][h0 + lr] = acc[i];
        }
        __syncthreads();

        // ia_sum = relu(LN_h(tques + img_embed)); iraw = ia_sum @ wia
        for (int r = w; r < 16; r += 8) {
            int n = m0 + r;
            if (n < CN) {
                float vv[CH / 32];
                float s = 0.f, sq = 0.f;
                #pragma unroll
                for (int i = 0; i < CH / 32; i++) {
                    int h = lane + 32 * i;
                    float v = tile[r][h] + ie[(size_t)n * CH + h];
                    vv[i] = v; s += v; sq += v * v;
                }
                s = wredsum(s); sq = wredsum(sq);
                float mu = s / CH, var = sq / CH - mu * mu, rs = rsqrtf(var + EPS);
                float acc = 0.f;
                #pragma unroll
                for (int i = 0; i < CH / 32; i++) {
                    int h = lane + 32 * i;
                    float v = fmaxf((vv[i] - mu) * rs * g4[h] + b4[h], 0.0f);
                    acc += v * wia[h];
                }
                acc = wredsum(acc);
                if (lane == 0) s_iraw[n] = acc + bia[0];
            }
        }
        __syncthreads();
    }

    // ---- P7: img_atten = softmax(LN_n(iraw, g5, b5)) (wave 0) -----------
    if (w == 0) {
        float x[4], y[4], e[4];
        float s = 0.f, sq = 0.f;
        #pragma unroll
        for (int i = 0; i < 4; i++) {
            int n = lane + 32 * i;
            x[i] = (n < CN) ? s_iraw[n] : 0.0f;
            s += x[i]; sq += x[i] * x[i];
        }
        s = wredsum(s); sq = wredsum(sq);
        float mu = s / CN, var = sq / CN - mu * mu, rs = rsqrtf(var + EPS);
        float m = -1e30f;
        #pragma unroll
        for (int i = 0; i < 4; i++) {
            int n = lane + 32 * i;
            y[i] = (n < CN) ? ((x[i] - mu) * rs * g5[n] + b5[n]) : -1e30f;
            m = fmaxf(m, y[i]);
        }
        m = wredmax(m);
        float se = 0.f;
        #pragma unroll
        for (int i = 0; i < 4; i++) {
            int n = lane + 32 * i;
            e[i] = (n < CN) ? __expf(y[i] - m) : 0.0f;
            se += e[i];
        }
        se = wredsum(se);
        #pragma unroll
        for (int i = 0; i < 4; i++) {
            int n = lane + 32 * i;
            if (n < CN) s_iatt[n] = e[i] / se;
        }
    }
    __syncthreads();

    // ---- P8: attention-weighted feature sums (float4-vectorized) --------
    // img_atten_feat: [D]; 2048/4 = 512 float4, 2 per thread
    #pragma unroll
    for (int c = 0; c < CD / 1024; c++) {
        int d4 = (tid + c * 256) * 4;
        float4 acc = { 0.f, 0.f, 0.f, 0.f };
        for (int n = 0; n < CN; n++) {
            float a = s_iatt[n];
            float4 v = *(const float4*)(imf + (size_t)n * CD + d4);
            acc.x += a * v.x; acc.y += a * v.y;
            acc.z += a * v.z; acc.w += a * v.w;
        }
        *(float4*)(out_img + (size_t)b * CD + d4) = acc;
    }
    // ques_atten_feat: [H]; 768/4 = 192 float4
    if (tid < CH / 4) {
        int h4 = tid * 4;
        float4 acc = { 0.f, 0.f, 0.f, 0.f };
        #pragma unroll
        for (int t = 0; t < CT; t++) {
            float a = s_qatt[t];
            float4 v = *(const float4*)(qf + (size_t)t * CH + h4);
            acc.x += a * v.x; acc.y += a * v.y;
            acc.z += a * v.z; acc.w += a * v.w;
        }
        *(float4*)(out_q + (size_t)b * CH + h4) = acc;
    }
}

// =====================================================================
extern "C" void kernel_launch(void* const* d_in, const int* in_sizes, int n_in,
                              void* d_out, int out_size, void* d_ws, size_t ws_size,
                              hipStream_t stream)
{
    const float* img_feat  = (const float*)d_in[0];
    const float* ques_feat = (const float*)d_in[1];
    const float* Wc  = (const float*)d_in[2];
    const float* bc  = (const float*)d_in[3];
    const float* Wq  = (const float*)d_in[4];
    const float* bq  = (const float*)d_in[5];
    const float* Wi  = (const float*)d_in[6];
    const float* bi  = (const float*)d_in[7];
    const float* wqa = (const float*)d_in[8];
    const float* bqa = (const float*)d_in[9];
    const float* wia = (const float*)d_in[10];
    const float* bia = (const float*)d_in[11];
    const float* g1 = (const float*)d_in[12];
    const float* b1 = (const float*)d_in[13];
    const float* g2 = (const float*)d_in[14];
    const float* b2 = (const float*)d_in[15];
    const float* g3 = (const float*)d_in[16];
    const float* b3 = (const float*)d_in[17];
    const float* g4 = (const float*)d_in[18];
    const float* b4 = (const float*)d_in[19];
    const float* g5 = (const float*)d_in[20];
    const float* b5 = (const float*)d_in[21];

    const int B = CB, N = CN, T = CT, K = CD, H = CH;
    const int M1 = B * N;   // 51200
    const int M2 = B * T;   // 7168

    float* ws        = (float*)d_ws;
    float* img_corr  = ws;
    float* img_embed = img_corr  + (size_t)M1 * H;
    float* ques_emb  = img_embed + (size_t)M1 * H;

    float* out_img = (float*)d_out;            // [B, D] first (tuple order)
    float* out_q   = out_img + (size_t)B * K;  // [B, H]

    // img_corr = img_feat@Wc+bc ; img_embed = img_feat@Wi+bi (A read once)
    gemm_bias_kernel<true><<<dim3(M1 / 64, H / 64), 256, 0, stream>>>(
        img_feat, Wc, bc, Wi, bi, img_corr, img_embed, M1, K, H);

    // ques_embed = ques_feat@Wq+bq
    gemm_bias_kernel<false><<<dim3(M2 / 64, H / 64), 256, 0, stream>>>(
        ques_feat, Wq, bq, nullptr, nullptr, ques_emb, nullptr, M2, H, H);

    coattn_fused_kernel<<<dim3(B), 256, 0, stream>>>(
        img_feat, ques_feat, img_corr, img_embed, ques_emb,
        wqa, bqa, wia, bia,
        g1, b1, g2, b2, g3, b3, g4, b4, g5, b5,
        out_img, out_q);
}